// TrilinearInterpolateFeatures_42597485641913
// MI455X (gfx1250) — compile-verified
//
#include <hip/hip_runtime.h>
#include <hip/hip_bf16.h>
#include <stdint.h>

// Stage gathered feature rows through LDS with the CDNA5 async copy engine
// (GLOBAL_LOAD_ASYNC_TO_LDS_B128, tracked on ASYNCcnt).
#define USE_ASYNC_LDS 1

typedef float vf4 __attribute__((ext_vector_type(4)));  // native vector for NT ld/st

namespace {
constexpr int BATCH = 2;      // B (fixed by reference)
constexpr int GD    = 128;    // D (fixed by reference)
constexpr int CH    = 128;    // C (fixed by reference)
constexpr int WPB   = 8;      // waves (queries) per 256-thread block
constexpr int OCC_ELEMS = BATCH * GD * GD * GD;  // 4,194,304 ints = 16 MB
}

// ---------------------------------------------------------------- init ----
__global__ __launch_bounds__(256) void init_ws_kernel(int* __restrict__ occ,
                                                      int* __restrict__ shift,
                                                      float* __restrict__ accum,
                                                      int n_vox) {
  int i = blockIdx.x * blockDim.x + threadIdx.x;
  if (i < OCC_ELEMS) occ[i] = -1;
  if (i < BATCH * 3) shift[i] = 0x7fffffff;
  if (i < n_vox) accum[i] = 0.0f;
}

// ------------------------------------------------------- per-batch min ----
__global__ __launch_bounds__(256) void shift_min_kernel(const int* __restrict__ coords,
                                                        int* __restrict__ shift,
                                                        int n_vox) {
  __shared__ int smin[BATCH * 3];
  if (threadIdx.x < BATCH * 3) smin[threadIdx.x] = 0x7fffffff;
  __syncthreads();
  int n = blockIdx.x * blockDim.x + threadIdx.x;
  if (n < n_vox) {
    int4 c = reinterpret_cast<const int4*>(coords)[n];  // (b, x, y, z)
    atomicMin(&smin[c.x * 3 + 0], c.y);
    atomicMin(&smin[c.x * 3 + 1], c.z);
    atomicMin(&smin[c.x * 3 + 2], c.w);
  }
  __syncthreads();
  if (threadIdx.x < BATCH * 3) atomicMin(&shift[threadIdx.x], smin[threadIdx.x]);
}

// --------------------------------------------------------- occ scatter ----
__global__ __launch_bounds__(256) void scatter_occ_kernel(const int* __restrict__ coords,
                                                          const int* __restrict__ shift,
                                                          int* __restrict__ occ,
                                                          int n_vox) {
  int n = blockIdx.x * blockDim.x + threadIdx.x;
  if (n >= n_vox) return;
  int4 c = reinterpret_cast<const int4*>(coords)[n];
  int sx = c.y - shift[c.x * 3 + 0];
  int sy = c.z - shift[c.x * 3 + 1];
  int sz = c.w - shift[c.x * 3 + 2];
  occ[((c.x * GD + sx) * GD + sy) * GD + sz] = n;
}

// ------------------------------------------------------------- queries ----
// One wave32 per query: lanes 0..7 own the 8 cube corners; all 32 lanes then
// each own 4 channels (lane*4 .. lane*4+3) of the C=128 feature blend.
// Pipeline: occ lookups -> issue 8 async row copies -> (overlap: index/wv
// stores, accum atomics, wsum shuffle reduction) -> wait ASYNCcnt -> FMA.
__global__ __launch_bounds__(256) void query_kernel(
    const float* __restrict__ qpts, const float* __restrict__ feat,
    const int* __restrict__ occ, const int* __restrict__ shift,
    float* __restrict__ out_qf, int* __restrict__ out_idx,
    float* __restrict__ out_wv, float* __restrict__ out_accum, int n_query) {
#if USE_ASYNC_LDS
  __shared__ float lbuf[WPB][8 * CH];  // 8 rows * 512 B per wave = 32 KB/WG
#endif
  const int lane  = threadIdx.x & 31;
  const int wslot = threadIdx.x >> 5;
  const int m = blockIdx.x * WPB + wslot;
  if (m >= n_query) return;  // whole wave exits together

  // streamed once -> non-temporal load keeps L2 for feat/occ working set
  const vf4 q4 = __builtin_nontemporal_load(
      reinterpret_cast<const vf4*>(qpts) + m);
  const int qb = (int)q4.x;
  const float qx = q4.y - (float)shift[qb * 3 + 0];
  const float qy = q4.z - (float)shift[qb * 3 + 1];
  const float qz = q4.w - (float)shift[qb * 3 + 2];
  const float fx0 = floorf(qx), fy0 = floorf(qy), fz0 = floorf(qz);
  const float fx = qx - fx0, fy = qy - fy0, fz = qz - fz0;
  const int x0 = (int)fx0, y0 = (int)fy0, z0 = (int)fz0;

  float wv_l = 0.0f;
  int idx_l = -1;
  bool valid = false;
  if (lane < 8) {
    const int oi = (lane >> 2) & 1, oj = (lane >> 1) & 1, ok = lane & 1;
    const int cx = x0 + oi, cy = y0 + oj, cz = z0 + ok;
    const bool inb = (cx >= 0) & (cx < GD) & (cy >= 0) & (cy < GD) &
                     (cz >= 0) & (cz < GD);
    const int ccx = min(max(cx, 0), GD - 1);
    const int ccy = min(max(cy, 0), GD - 1);
    const int ccz = min(max(cz, 0), GD - 1);
    const int vid = occ[((qb * GD + ccx) * GD + ccy) * GD + ccz];
    valid = inb && (vid >= 0);
    const float w = (oi ? fx : 1.0f - fx) * (oj ? fy : 1.0f - fy) *
                    (ok ? fz : 1.0f - fz);
    wv_l  = valid ? w : 0.0f;
    idx_l = valid ? vid : -1;
  }

  // broadcast corner (index, weight) to all lanes; invalid -> row 0, weight 0
  int vidk[8]; float wk[8];
#pragma unroll
  for (int k = 0; k < 8; ++k) {
    vidk[k] = max(__shfl(idx_l, k, 32), 0);
    wk[k]   = __shfl(wv_l, k, 32);
  }

#if USE_ASYNC_LDS
  // Issue all 8 row copies on the async engine ASAP (ASYNCcnt).
  // Each async b128 op: 32 lanes x 16 B = one full 512 B feature row.
  float* mybuf = &lbuf[wslot][0];
#pragma unroll
  for (int k = 0; k < 8; ++k) {
    const float* g = feat + (size_t)vidk[k] * CH + lane * 4;
    unsigned lds_off =
        (unsigned)(uintptr_t)(&mybuf[k * CH + lane * 4]);  // low 32b = LDS offset
    unsigned long long ga = (unsigned long long)(uintptr_t)g;
    asm volatile("global_load_async_to_lds_b128 %0, %1, off"
                 :: "v"(lds_off), "v"(ga) : "memory");
  }
#endif

  // ---- overlap region: independent outputs while the DMA fills LDS ----
  if (lane < 8) {
    __builtin_nontemporal_store(idx_l, out_idx + (size_t)m * 8 + lane);
    __builtin_nontemporal_store(wv_l,  out_wv  + (size_t)m * 8 + lane);
    if (valid) atomicAdd(&out_accum[idx_l], wv_l);  // segment_sum; invalid adds 0
  }

  // wsum across the 8 corner lanes (wave32 xor-shuffle tree, lanes>=8 hold 0)
  float r = wv_l;
  r += __shfl_xor(r, 1, 32);
  r += __shfl_xor(r, 2, 32);
  r += __shfl_xor(r, 4, 32);
  const float wsum = __shfl(r, 0, 32);
  const float inv = 1.0f / fmaxf(wsum, 1e-8f);

  vf4 acc = {0.f, 0.f, 0.f, 0.f};
#if USE_ASYNC_LDS
  asm volatile("s_wait_asynccnt 0" ::: "memory");  // per-wave ASYNCcnt drain
#pragma unroll
  for (int k = 0; k < 8; ++k) {
    const vf4 fr = *reinterpret_cast<const vf4*>(&mybuf[k * CH + lane * 4]);
    acc += wk[k] * fr;
  }
#else
  vf4 fr[8];
#pragma unroll
  for (int k = 0; k < 8; ++k)  // 8 coalesced b128 loads in flight
    fr[k] = reinterpret_cast<const vf4*>(feat)[(size_t)vidk[k] * (CH / 4) + lane];
#pragma unroll
  for (int k = 0; k < 8; ++k) acc += wk[k] * fr[k];
#endif

  const vf4 o = acc * inv;
  // written once, never re-read by this dispatch -> non-temporal
  __builtin_nontemporal_store(
      o, reinterpret_cast<vf4*>(out_qf) + (size_t)m * (CH / 4) + lane);
}

// -------------------------------------------------------------- launch ----
extern "C" void kernel_launch(void* const* d_in, const int* in_sizes, int n_in,
                              void* d_out, int out_size, void* d_ws, size_t ws_size,
                              hipStream_t stream) {
  const int*   coords = (const int*)d_in[0];
  const float* feats  = (const float*)d_in[1];
  const float* qpts   = (const float*)d_in[2];
  const int n_vox   = in_sizes[0] / 4;
  const int n_query = in_sizes[2] / 4;

  // workspace: [occ grid 16 MB][shift 6 ints]
  int* occ   = (int*)d_ws;
  int* shift = occ + (size_t)OCC_ELEMS;

  // d_out: query_features | interpolation_indices | wv | accum_voxel_weights
  float* out_qf    = (float*)d_out;
  int*   out_idx   = (int*)(out_qf + (size_t)n_query * CH);
  float* out_wv    = (float*)(out_idx + (size_t)n_query * 8);
  float* out_accum = out_wv + (size_t)n_query * 8;

  const int tb = 256;
  hipLaunchKernelGGL(init_ws_kernel, dim3((OCC_ELEMS + tb - 1) / tb), dim3(tb),
                     0, stream, occ, shift, out_accum, n_vox);
  hipLaunchKernelGGL(shift_min_kernel, dim3((n_vox + tb - 1) / tb), dim3(tb),
                     0, stream, coords, shift, n_vox);
  hipLaunchKernelGGL(scatter_occ_kernel, dim3((n_vox + tb - 1) / tb), dim3(tb),
                     0, stream, coords, shift, occ, n_vox);
  hipLaunchKernelGGL(query_kernel, dim3((n_query + WPB - 1) / WPB), dim3(tb),
                     0, stream, qpts, feats, occ, shift,
                     out_qf, out_idx, out_wv, out_accum, n_query);
}